// SynthesisBLK_1340029796571
// MI455X (gfx1250) — compile-verified
//
#include <hip/hip_runtime.h>
#include <hip/hip_bf16.h>

typedef __attribute__((ext_vector_type(2))) float v2f;
typedef __attribute__((ext_vector_type(8))) float v8f;

#define BATCH 16
#define CIN   256
#define COUT  256
#define HDIM  64
#define NPIX  4096           // 64*64
#define WSTY  512

// workspace layout (float offsets)
#define WS_Y      ((size_t)0)                       // 16*256*4096 = 16,777,216 floats (67 MB)
#define WS_WB     ((size_t)16777216)                // 16*256*256  =  1,048,576 floats (4 MB)
#define WS_STYLES ((size_t)(16777216 + 1048576))    // 4096 floats
#define WS_SSQ    (WS_STYLES + 4096)                // 1 float

// ---------------------------------------------------------------------------
__global__ void k_init(float* ws) {
    if (threadIdx.x == 0) ws[WS_SSQ] = 0.0f;
}

// styles[b,i] = (1/sqrt(512)) * dot(w[b,:], affine_w[i,:]) + affine_b[i]
// also accumulate sum(styles^2) globally.
__global__ __launch_bounds__(256) void k_styles(const float* __restrict__ w,
                                                const float* __restrict__ aw,
                                                const float* __restrict__ ab,
                                                float* __restrict__ ws) {
    __shared__ float wrow[WSTY];
    __shared__ float red[256];
    const int b = blockIdx.x;
    const int i = threadIdx.x;
    wrow[i]       = w[b * WSTY + i];
    wrow[i + 256] = w[b * WSTY + i + 256];
    __syncthreads();
    const float wg = 0.044194173824159216f;  // 1/sqrt(512)
    const float* awr = aw + (size_t)i * WSTY;
    float acc = 0.0f;
#pragma unroll 8
    for (int k = 0; k < WSTY; ++k) acc += wrow[k] * awr[k];
    const float st = acc * wg + ab[i];
    ws[WS_STYLES + b * 256 + i] = st;
    red[i] = st * st;
    __syncthreads();
    for (int s = 128; s > 0; s >>= 1) {
        if (i < s) red[i] += red[i + s];
        __syncthreads();
    }
    if (i == 0) atomicAdd(&ws[WS_SSQ], red[0]);
}

// wb[b,o,i] = demodulated modulated weight; one block per (b,o) pair.
__global__ __launch_bounds__(256) void k_wb(const float* __restrict__ weight,
                                            float* __restrict__ ws) {
    __shared__ float red[256];
    const int bo = blockIdx.x;          // b*256 + o
    const int b = bo >> 8, o = bo & 255;
    const int i = threadIdx.x;
    const float wv = weight[(size_t)o * CIN + i];
    red[i] = wv * wv;
    __syncthreads();
    for (int s = 128; s > 0; s >>= 1) {
        if (i < s) red[i] += red[i + s];
        __syncthreads();
    }
    const float wnorm = rsqrtf(red[0] * (1.0f / 256.0f));
    __syncthreads();
    const float srs = rsqrtf(ws[WS_SSQ] * (1.0f / 4096.0f));
    const float sv  = ws[WS_STYLES + b * 256 + i] * srs;
    const float wbv = wv * wnorm * sv;
    red[i] = wbv * wbv;
    __syncthreads();
    for (int s = 128; s > 0; s >>= 1) {
        if (i < s) red[i] += red[i + s];
        __syncthreads();
    }
    const float d = rsqrtf(red[0] + 1e-8f);
    ws[WS_WB + (size_t)bo * CIN + i] = wbv * d;
}

// ---------------------------------------------------------------------------
// Main GEMM: for each b, Y[o,n] = sum_i wb[b,o,i] * x[b,i,n] + bias[o]
// Block tile 64(M=o) x 64(N=n), K=256 in 16-wide double-buffered LDS steps.
// 8 waves (wave32): wave = (wm in 0..1, wn in 0..3); each wave owns two
// 16x16 f32 accumulators (rows wm*32+{0,16}, cols wn*16).
#define MT 64
#define NT 64
#define KT 16
#define LDT 20   // padded LDS row stride (floats): conflict-free, 16B multiple

__global__ __launch_bounds__(256) void k_gemm(const float* __restrict__ x,
                                              const float* __restrict__ bias,
                                              float* __restrict__ ws) {
    __shared__ __align__(16) float As[2][MT * LDT];  // [m][k]
    __shared__ __align__(16) float Bs[2][NT * LDT];  // [n][k]

    const int b  = blockIdx.z;
    const int mo = blockIdx.y * MT;
    const int no = blockIdx.x * NT;
    const int t  = threadIdx.x;
    const int lane  = t & 31;
    const int wave  = t >> 5;
    const int wn    = wave & 3;
    const int wm    = wave >> 2;
    const int lhalf = lane >> 4;   // 0: lanes 0-15, 1: lanes 16-31
    const int lm    = lane & 15;

    const float* xb  = x  + (size_t)b * CIN * NPIX;
    const float* wbb = ws + WS_WB + (size_t)b * COUT * CIN;

    // cooperative-load indices (256 threads, one float4 of A and B each)
    const int la_m = t >> 2;           // 0..63  (o row)
    const int la_k = (t & 3) * 4;      // 0,4,8,12
    const int lb_k = t >> 4;           // 0..15  (k row)
    const int lb_n = (t & 15) * 4;     // 0..60  (n group)

    v8f c0 = {0.f, 0.f, 0.f, 0.f, 0.f, 0.f, 0.f, 0.f};
    v8f c1 = c0;

    // prologue: stage K-tile 0
    {
        const float4 av = *(const float4*)(wbb + (size_t)(mo + la_m) * CIN + la_k);
        *(float4*)(&As[0][la_m * LDT + la_k]) = av;
        const float4 bv = *(const float4*)(xb + (size_t)lb_k * NPIX + no + lb_n);
        Bs[0][(lb_n + 0) * LDT + lb_k] = bv.x;
        Bs[0][(lb_n + 1) * LDT + lb_k] = bv.y;
        Bs[0][(lb_n + 2) * LDT + lb_k] = bv.z;
        Bs[0][(lb_n + 3) * LDT + lb_k] = bv.w;
    }
    __syncthreads();

    const int m0 = wm * 32;
    const int n0 = wn * 16;

    for (int kt = 0; kt < CIN / KT; ++kt) {
        const int cur = kt & 1;
        const int nxt = cur ^ 1;
        if (kt + 1 < CIN / KT) {
            const int k0 = (kt + 1) * KT;
            const float4 av = *(const float4*)(wbb + (size_t)(mo + la_m) * CIN + k0 + la_k);
            *(float4*)(&As[nxt][la_m * LDT + la_k]) = av;
            const float4 bv = *(const float4*)(xb + (size_t)(k0 + lb_k) * NPIX + no + lb_n);
            Bs[nxt][(lb_n + 0) * LDT + lb_k] = bv.x;
            Bs[nxt][(lb_n + 1) * LDT + lb_k] = bv.y;
            Bs[nxt][(lb_n + 2) * LDT + lb_k] = bv.z;
            Bs[nxt][(lb_n + 3) * LDT + lb_k] = bv.w;
        }
        const float* Ab = &As[cur][0];
        const float* Bb = &Bs[cur][0];
#pragma unroll
        for (int ks = 0; ks < 4; ++ks) {
            // A frag (16x4 f32): lanes 0-15 hold K={base,base+1}, lanes 16-31 K={base+2,base+3}
            const int kk = ks * 4 + lhalf * 2;
            const v2f bfrag = *(const v2f*)(Bb + (n0 + lm) * LDT + kk);
            const v2f a0    = *(const v2f*)(Ab + (m0 + lm) * LDT + kk);
            const v2f a1    = *(const v2f*)(Ab + (m0 + 16 + lm) * LDT + kk);
            c0 = __builtin_amdgcn_wmma_f32_16x16x4_f32(false, a0, false, bfrag,
                                                       (short)0, c0, false, false);
            c1 = __builtin_amdgcn_wmma_f32_16x16x4_f32(false, a1, false, bfrag,
                                                       (short)0, c1, false, false);
        }
        __syncthreads();
    }

    // epilogue: C 16x16 f32 layout — VGPR r: M = r + 8*lhalf, N = lm
    float* y = ws + WS_Y + (size_t)b * COUT * NPIX;
    const int ncol = no + n0 + lm;
#pragma unroll
    for (int r = 0; r < 8; ++r) {
        const int o0 = mo + m0 + r + 8 * lhalf;
        y[(size_t)o0 * NPIX + ncol] = c0[r] + bias[o0];
        const int o1 = o0 + 16;
        y[(size_t)o1 * NPIX + ncol] = c1[r] + bias[o1];
    }
}

// ---------------------------------------------------------------------------
// 2x bilinear upsample (half-pixel centers): out coord j maps to j/2 - 0.25.
// Even j: ih0=j/2-1, w1=0.75 ; Odd j: ih0=j/2, w1=0.25. Edge-clamped (matches
// jax renormalized linear kernel at borders).
__global__ __launch_bounds__(256) void k_upsample(const float* __restrict__ ws,
                                                  float* __restrict__ out) {
    const float* y = ws + WS_Y;
    const size_t idx = (size_t)blockIdx.x * 256 + threadIdx.x;
    const int ow = (int)(idx & 127);
    const int oh = (int)((idx >> 7) & 127);
    const size_t c = idx >> 14;  // b*256 + o
    const float* yc = y + c * NPIX;

    int ih0 = (oh - 1) >> 1;
    const float fh = (oh & 1) ? 0.25f : 0.75f;
    int iw0 = (ow - 1) >> 1;
    const float fw = (ow & 1) ? 0.25f : 0.75f;
    const int ih1 = min(ih0 + 1, HDIM - 1); ih0 = max(ih0, 0);
    const int iw1 = min(iw0 + 1, HDIM - 1); iw0 = max(iw0, 0);

    const float v00 = yc[ih0 * HDIM + iw0];
    const float v01 = yc[ih0 * HDIM + iw1];
    const float v10 = yc[ih1 * HDIM + iw0];
    const float v11 = yc[ih1 * HDIM + iw1];
    const float top = v00 + fw * (v01 - v00);
    const float bot = v10 + fw * (v11 - v10);
    out[idx] = top + fh * (bot - top);
}

// ---------------------------------------------------------------------------
extern "C" void kernel_launch(void* const* d_in, const int* in_sizes, int n_in,
                              void* d_out, int out_size, void* d_ws, size_t ws_size,
                              hipStream_t stream) {
    const float* x      = (const float*)d_in[0];  // (16,256,64,64)
    const float* w      = (const float*)d_in[1];  // (16,512)
    const float* aw     = (const float*)d_in[2];  // (256,512)
    const float* ab     = (const float*)d_in[3];  // (256,)
    const float* weight = (const float*)d_in[4];  // (256,256,1,1)
    const float* bias   = (const float*)d_in[5];  // (256,)
    float* ws  = (float*)d_ws;
    float* out = (float*)d_out;

    k_init<<<1, 32, 0, stream>>>(ws);
    k_styles<<<BATCH, 256, 0, stream>>>(w, aw, ab, ws);
    k_wb<<<BATCH * COUT, 256, 0, stream>>>(weight, ws);
    dim3 g(NPIX / NT, COUT / MT, BATCH);      // (64, 4, 16)
    k_gemm<<<g, 256, 0, stream>>>(x, bias, ws);
    k_upsample<<<(BATCH * COUT * 4 * NPIX) / 256, 256, 0, stream>>>(ws, out);
}